// RetinaNet_19164144075109
// MI455X (gfx1250) — compile-verified
//
#include <hip/hip_runtime.h>
#include <stdint.h>

#define SCORE_THRESH 0.05f
#define IOU_THRESH   0.5f
#define TOPK         1000
#define MAXDET       100
#define NBINS        65536
#define CAND_MAX     4096
#define K1_THREADS   128
#define STRIDEF      84     // padded floats per anchor row in LDS (336B, 16B aligned)

// ---------------- CDNA5 async global->LDS copy (ASYNCcnt path) ---------------
__device__ __forceinline__ void async_load_b128(uint32_t lds_off, const void* gaddr) {
    // vdst = LDS byte offset (VGPR), vaddr = 64-bit global address (VGPR pair)
    asm volatile("global_load_async_to_lds_b128 %0, %1, off"
                 :: "v"(lds_off), "v"(gaddr) : "memory");
}
__device__ __forceinline__ void wait_async0() {
    asm volatile("s_wait_asynccnt 0" ::: "memory");
}

// ---------------- kernel 0: reset scratch state (runs every call) ------------
__global__ void k_zero(int* __restrict__ hist, int* __restrict__ counter,
                       int* __restrict__ cutoff) {
    int i = blockIdx.x * blockDim.x + threadIdx.x;
    if (i < NBINS) hist[i] = 0;
    if (i == 0) { *counter = 0; *cutoff = NBINS; }
}

// ---------------- kernel 1: streaming max/argmax via async LDS staging -------
__global__ void k_score_hist(const float* __restrict__ cls,   // [A,C], C==80
                             float* __restrict__ scoreOut,    // [A] masked score
                             int*   __restrict__ clsOut,      // [A] argmax
                             int*   __restrict__ hist,
                             int A, int C) {
    __shared__ __align__(16) float buf[K1_THREADS * STRIDEF];
    const int t = threadIdx.x;
    const int anchor = blockIdx.x * K1_THREADS + t;
    const int srcAnchor = (anchor < A) ? anchor : 0;           // tail lanes: dup load
    const char* src = (const char*)(cls + (size_t)srcAnchor * C);
    const uint32_t ldsOff = (uint32_t)(uintptr_t)(&buf[t * STRIDEF]);

    const int nvec = (C + 3) >> 2;                              // 20 x b128 per lane
    for (int k = 0; k < nvec; ++k)
        async_load_b128(ldsOff + (uint32_t)(k * 16), src + k * 16);
    wait_async0();
    __syncthreads();

    if (anchor < A) {
        const float* row = &buf[t * STRIDEF];
        float best = -1e30f; int bi = 0;
        for (int c = 0; c < C; ++c) {
            float v = row[c];
            if (v > best) { best = v; bi = c; }
        }
        clsOut[anchor] = bi;
        float ms = (best > SCORE_THRESH) ? best : -1.0f;
        scoreOut[anchor] = ms;
        if (ms > 0.0f) {
            int bin = (int)(ms * (float)NBINS);
            bin = bin < 0 ? 0 : (bin >= NBINS ? NBINS - 1 : bin);
            atomicAdd(&hist[bin], 1);
        }
    }
}

// Fallback without the 80-class LDS tiling (generic C)
__global__ void k_score_plain(const float* __restrict__ cls,
                              float* __restrict__ scoreOut, int* __restrict__ clsOut,
                              int* __restrict__ hist, int A, int C) {
    int anchor = blockIdx.x * blockDim.x + threadIdx.x;
    if (anchor >= A) return;
    const float* row = cls + (size_t)anchor * C;
    float best = -1e30f; int bi = 0;
    for (int c = 0; c < C; ++c) { float v = row[c]; if (v > best) { best = v; bi = c; } }
    clsOut[anchor] = bi;
    float ms = (best > SCORE_THRESH) ? best : -1.0f;
    scoreOut[anchor] = ms;
    if (ms > 0.0f) {
        int bin = (int)(ms * (float)NBINS);
        bin = bin < 0 ? 0 : (bin >= NBINS ? NBINS - 1 : bin);
        atomicAdd(&hist[bin], 1);
    }
}

// ---------------- kernel 2: find exact top-1000 cutoff bin -------------------
__global__ void k_scan(const int* __restrict__ hist, int* __restrict__ cutoff) {
    __shared__ int part[1024];
    const int t = threadIdx.x;
    const int chunk = NBINS / 1024;                 // 64 bins per thread, descending
    const int base = NBINS - (t + 1) * chunk;
    int s = 0;
    for (int i = 0; i < chunk; ++i) s += hist[base + i];
    part[t] = s;
    __syncthreads();
    for (int off = 1; off < 1024; off <<= 1) {      // inclusive scan (cum from top)
        int v = part[t];
        if (t >= off) v += part[t - off];
        __syncthreads();
        part[t] = v;
        __syncthreads();
    }
    if (t == 0) {
        int total = part[1023];
        int K = total < TOPK ? total : TOPK;
        if (K <= 0) { *cutoff = NBINS; return; }
        int ts = 0;
        while (part[ts] < K) ++ts;
        int acc = (ts > 0) ? part[ts - 1] : 0;
        int b = NBINS - ts * chunk - 1;
        while (b >= 0) {
            acc += hist[b];
            if (acc >= K) break;
            --b;
        }
        *cutoff = (b < 0) ? 0 : b;
    }
}

// ---------------- kernel 3: compact candidates at/above cutoff ---------------
__global__ void k_compact(const float* __restrict__ scores, int A,
                          const int* __restrict__ cutoff, int* __restrict__ counter,
                          float* __restrict__ candS, int* __restrict__ candI) {
    int i = blockIdx.x * blockDim.x + threadIdx.x;
    if (i >= A) return;
    float s = scores[i];
    if (s <= SCORE_THRESH) return;
    int bin = (int)(s * (float)NBINS);
    bin = bin < 0 ? 0 : (bin >= NBINS ? NBINS - 1 : bin);
    if (bin >= *cutoff) {
        int pos = atomicAdd(counter, 1);
        if (pos < CAND_MAX) { candS[pos] = s; candI[pos] = i; }
    }
}

// ---------------- kernel 4: sort + decode + greedy NMS + outputs -------------
__global__ void k_sort_nms(const float* __restrict__ candS, const int* __restrict__ candI,
                           const int* __restrict__ counter,
                           const float* __restrict__ regr,   // [A,4]
                           const float* __restrict__ anch,   // [A,4]
                           const int*   __restrict__ clsArr, // [A]
                           const int* __restrict__ rawH, const int* __restrict__ rawW,
                           float* __restrict__ out) {        // 600 floats
    __shared__ float key[CAND_MAX];
    __shared__ int   idx[CAND_MAX];
    __shared__ float bx[TOPK * 4];
    __shared__ float area[TOPK];
    __shared__ int   keep[TOPK];
    __shared__ int   sel[MAXDET];
    const int t = threadIdx.x;

    int n = *counter; if (n > CAND_MAX) n = CAND_MAX;
    for (int i = t; i < CAND_MAX; i += 1024) {
        if (i < n) { key[i] = candS[i]; idx[i] = candI[i]; }
        else       { key[i] = -1.0f;    idx[i] = 0x7FFFFFFF; }
    }
    __syncthreads();

    // bitonic sort: score desc, tie -> lower anchor index first (top_k stability)
    for (int k = 2; k <= CAND_MAX; k <<= 1) {
        for (int j = k >> 1; j > 0; j >>= 1) {
            for (int i = t; i < CAND_MAX; i += 1024) {
                int ixj = i ^ j;
                if (ixj > i) {
                    float a = key[i], b = key[ixj];
                    int ia = idx[i], ib = idx[ixj];
                    bool aFirst = (a > b) || (a == b && ia < ib);
                    bool desc = ((i & k) == 0);
                    if (desc ? !aFirst : aFirst) {
                        key[i] = b; key[ixj] = a;
                        idx[i] = ib; idx[ixj] = ia;
                    }
                }
            }
            __syncthreads();
        }
    }

    // image dims: accept int32 or float32 scalar encoding
    int rw = *rawW, rh = *rawH;
    float imgW = (rw > 0 && rw < (1 << 24)) ? (float)rw : __int_as_float(rw);
    float imgH = (rh > 0 && rh < (1 << 24)) ? (float)rh : __int_as_float(rh);

    // decode + clip boxes for the TOPK candidates only
    for (int i = t; i < TOPK; i += 1024) {
        float s = key[i];
        float x1 = 0, y1 = 0, x2 = 0, y2 = 0;
        int kp = 0;
        if (s > SCORE_THRESH) {
            int ai = idx[i];
            const float* an = anch + (size_t)ai * 4;
            const float* rg = regr + (size_t)ai * 4;
            float aw = an[2] - an[0], ah = an[3] - an[1];
            float acx = an[0] + 0.5f * aw, acy = an[1] + 0.5f * ah;
            float dx = rg[0] * 0.1f, dy = rg[1] * 0.1f;
            float dw = rg[2] * 0.2f, dh = rg[3] * 0.2f;
            float pcx = acx + dx * aw, pcy = acy + dy * ah;
            float pw = expf(dw) * aw, ph = expf(dh) * ah;
            x1 = fminf(fmaxf(pcx - 0.5f * pw, 0.f), imgW);
            y1 = fminf(fmaxf(pcy - 0.5f * ph, 0.f), imgH);
            x2 = fminf(fmaxf(pcx + 0.5f * pw, 0.f), imgW);
            y2 = fminf(fmaxf(pcy + 0.5f * ph, 0.f), imgH);
            kp = 1;
        }
        bx[i * 4 + 0] = x1; bx[i * 4 + 1] = y1;
        bx[i * 4 + 2] = x2; bx[i * 4 + 3] = y2;
        area[i] = fmaxf(x2 - x1, 0.f) * fmaxf(y2 - y1, 0.f);
        keep[i] = kp;
    }
    __syncthreads();

    // greedy NMS: row i suppresses j>i in parallel
    for (int i = 0; i < TOPK; ++i) {
        if (keep[i]) {
            float bx1 = bx[i * 4], by1 = bx[i * 4 + 1];
            float bx2 = bx[i * 4 + 2], by2 = bx[i * 4 + 3];
            float ar = area[i];
            int j = i + 1 + t;
            if (j < TOPK && keep[j]) {
                float xx1 = fmaxf(bx1, bx[j * 4]),     yy1 = fmaxf(by1, bx[j * 4 + 1]);
                float xx2 = fminf(bx2, bx[j * 4 + 2]), yy2 = fminf(by2, bx[j * 4 + 3]);
                float inter = fmaxf(xx2 - xx1, 0.f) * fmaxf(yy2 - yy1, 0.f);
                float iou = inter / (ar + area[j] - inter + 1e-8f);
                if (iou > IOU_THRESH) keep[j] = 0;
            }
        }
        __syncthreads();
    }

    if (t == 0) {
        int s = 0;
        for (int i = 0; i < TOPK && s < MAXDET; ++i)
            if (keep[i]) sel[s++] = i;
        for (; s < MAXDET; ++s) sel[s] = -1;
    }
    __syncthreads();

    if (t < MAXDET) {
        int i = sel[t];
        if (i >= 0) {
            out[t] = key[i];
            out[MAXDET + t] = (float)clsArr[idx[i]];
            out[2 * MAXDET + t * 4 + 0] = bx[i * 4 + 0];
            out[2 * MAXDET + t * 4 + 1] = bx[i * 4 + 1];
            out[2 * MAXDET + t * 4 + 2] = bx[i * 4 + 2];
            out[2 * MAXDET + t * 4 + 3] = bx[i * 4 + 3];
        } else {
            out[t] = 0.0f;
            out[MAXDET + t] = -1.0f;
            out[2 * MAXDET + t * 4 + 0] = 0.0f;
            out[2 * MAXDET + t * 4 + 1] = 0.0f;
            out[2 * MAXDET + t * 4 + 2] = 0.0f;
            out[2 * MAXDET + t * 4 + 3] = 0.0f;
        }
    }
}

extern "C" void kernel_launch(void* const* d_in, const int* in_sizes, int n_in,
                              void* d_out, int out_size, void* d_ws, size_t ws_size,
                              hipStream_t stream) {
    const float* cls  = (const float*)d_in[0];
    const float* regr = (const float*)d_in[1];
    const float* anch = (const float*)d_in[2];
    const int* rawH = (const int*)d_in[3];
    const int* rawW = (const int*)d_in[4];
    (void)n_in; (void)out_size; (void)ws_size;

    const int A = in_sizes[1] / 4;
    const int C = (A > 0) ? in_sizes[0] / A : 0;
    const int Apad = (A + 3) & ~3;

    char* w = (char*)d_ws;
    float* scores = (float*)w;  w += sizeof(float) * Apad;
    int* clsArr   = (int*)w;    w += sizeof(int) * Apad;
    int* hist     = (int*)w;    w += sizeof(int) * NBINS;
    int* counter  = (int*)w;    w += 16;
    int* cutoff   = (int*)w;    w += 16;
    float* candS  = (float*)w;  w += sizeof(float) * CAND_MAX;
    int* candI    = (int*)w;    w += sizeof(int) * CAND_MAX;

    k_zero<<<(NBINS + 255) / 256, 256, 0, stream>>>(hist, counter, cutoff);
    if (C == 80) {
        k_score_hist<<<(A + K1_THREADS - 1) / K1_THREADS, K1_THREADS, 0, stream>>>(
            cls, scores, clsArr, hist, A, C);
    } else {
        k_score_plain<<<(A + 255) / 256, 256, 0, stream>>>(cls, scores, clsArr, hist, A, C);
    }
    k_scan<<<1, 1024, 0, stream>>>(hist, cutoff);
    k_compact<<<(A + 255) / 256, 256, 0, stream>>>(scores, A, cutoff, counter, candS, candI);
    k_sort_nms<<<1, 1024, 0, stream>>>(candS, candI, counter, regr, anch, clsArr,
                                       rawH, rawW, (float*)d_out);
}